// UserAttentionPooling_30193620090917
// MI455X (gfx1250) — compile-verified
//
#include <hip/hip_runtime.h>
#include <math.h>

typedef __attribute__((ext_vector_type(2))) float v2f;
typedef __attribute__((ext_vector_type(8))) float v8f;

#define DIM 512
#define TILE 16

__device__ __forceinline__ float waveMax32(float v) {
#pragma unroll
  for (int m = 16; m > 0; m >>= 1) v = fmaxf(v, __shfl_xor(v, m, 32));
  return v;
}
__device__ __forceinline__ float waveSum32(float v) {
#pragma unroll
  for (int m = 16; m > 0; m >>= 1) v += __shfl_xor(v, m, 32);
  return v;
}
// SGPR broadcast of lane `l` (uniform, compile-time) -> v_readlane_b32
__device__ __forceinline__ float bcastLane(float v, int l) {
  return __builtin_bit_cast(float,
      __builtin_amdgcn_readlane(__builtin_bit_cast(int, v), l));
}
__device__ __forceinline__ void fma4(float4& a, float w, const float4 x) {
  a.x = fmaf(w, x.x, a.x); a.y = fmaf(w, x.y, a.y);
  a.z = fmaf(w, x.z, a.z); a.w = fmaf(w, x.w, a.w);
}
__device__ __forceinline__ void scale4(float4& a, float r) {
  a.x *= r; a.y *= r; a.z *= r; a.w *= r;
}

// ---------------------------------------------------------------------------
// Kernel 1: fused streaming pass. Each wave owns a (m, s, acc[512]) online-
// softmax state and grid-strides over 16-row tiles.
//   Phase A: scores of 16 rows via chained V_WMMA_F32_16X16X4_F32.
//            A = q chunk (broadcast, documented 16x4 layout),
//            B = 16 rows' chunk (column-major mirror layout).
//            C[i][j] = dot(q, row_j)  -> lane L holds score of row r0+(L&15).
//   Phase B: online softmax update + coalesced float4 accumulation with
//            per-row weights broadcast via v_readlane (SGPR operand in FMAs).
// ---------------------------------------------------------------------------
__global__ __launch_bounds__(256)
void uap_stream_kernel(const float* __restrict__ emb, const float* __restrict__ q,
                       float* __restrict__ accA, float* __restrict__ mArr,
                       float* __restrict__ sArr, int nTiles, int nWaves) {
  __shared__ float qs[DIM];
  const int tid = threadIdx.x;
  for (int i = tid; i < DIM; i += 256) qs[i] = q[i];
  __syncthreads();

  const int lane  = tid & 31;
  const int gwave = blockIdx.x * 8 + (tid >> 5);
  const int half  = lane >> 4;   // 0: K=0,1   1: K=2,3
  const int lsub  = lane & 15;   // row-in-tile (phase A B-operand)

  float4 acc0 = {0,0,0,0}, acc1 = {0,0,0,0}, acc2 = {0,0,0,0}, acc3 = {0,0,0,0};
  float m = -INFINITY;
  float s = 0.0f;

  const v2f*    qv   = (const v2f*)qs;
  const float4* emb4 = (const float4*)emb;

  for (int t = gwave; t < nTiles; t += nWaves) {
    const int r0 = t * TILE;

    // ---- Phase A: 16 scores via WMMA f32 16x16x4 (two independent chains) --
    const v2f* rowv = (const v2f*)(emb + (size_t)(r0 + lsub) * DIM);
    v8f c0 = {0,0,0,0,0,0,0,0};
    v8f c1 = {0,0,0,0,0,0,0,0};
#pragma unroll 4
    for (int kd = 0; kd < DIM; kd += 8) {
      const int i0 = (kd >> 1) + half;   // float2 index of K-chunk kd
      const int i1 = i0 + 2;             // K-chunk kd+4
      v2f a0 = qv[i0],   a1 = qv[i1];
      v2f b0 = rowv[i0], b1 = rowv[i1];
      c0 = __builtin_amdgcn_wmma_f32_16x16x4_f32(false, a0, false, b0,
                                                 (short)0, c0, false, false);
      c1 = __builtin_amdgcn_wmma_f32_16x16x4_f32(false, a1, false, b1,
                                                 (short)0, c1, false, false);
    }
    const float sLane = c0[0] + c1[0];   // score of row r0 + (lane & 15)

    // ---- Phase B: online softmax update --------------------------------
    const float mTile = waveMax32(sLane);
    const float mNew  = fmaxf(m, mTile);
    const float resc  = expf(m - mNew);  // first tile: exp(-inf)=0, acc/s are 0
    m = mNew;
    s *= resc;
    scale4(acc0, resc); scale4(acc1, resc); scale4(acc2, resc); scale4(acc3, resc);

    const float wLane = expf(sLane - m);       // weight of row r0+(lane&15)
    s += 0.5f * waveSum32(wLane);              // halves duplicate -> *0.5

#pragma unroll
    for (int j = 0; j < TILE; ++j) {
      const float wj = bcastLane(wLane, j);    // v_readlane_b32 -> SGPR
      const float4* rp = emb4 + (size_t)(r0 + j) * (DIM / 4) + lane;
      float4 x0 = rp[0], x1 = rp[32], x2 = rp[64], x3 = rp[96];
      fma4(acc0, wj, x0); fma4(acc1, wj, x1);
      fma4(acc2, wj, x2); fma4(acc3, wj, x3);
    }
  }

  // write per-wave partials: dim mapping d = (k*32+lane)*4 + c
  float4* outA = (float4*)accA;
  const size_t base = (size_t)gwave * (DIM / 4);
  outA[base +  0 + lane] = acc0;
  outA[base + 32 + lane] = acc1;
  outA[base + 64 + lane] = acc2;
  outA[base + 96 + lane] = acc3;
  if (lane == 0) { mArr[gwave] = m; sArr[gwave] = s; }
}

// ---------------------------------------------------------------------------
// Kernel 2a: global max M and sum S = sum_w s_w * exp(m_w - M)
// ---------------------------------------------------------------------------
__global__ __launch_bounds__(256)
void uap_ms_kernel(const float* __restrict__ mArr, const float* __restrict__ sArr,
                   float* __restrict__ MS, int W) {
  __shared__ float red[256];
  const int tid = threadIdx.x;
  float mv = -INFINITY;
  for (int w = tid; w < W; w += 256) mv = fmaxf(mv, mArr[w]);
  red[tid] = mv; __syncthreads();
  for (int o = 128; o > 0; o >>= 1) {
    if (tid < o) red[tid] = fmaxf(red[tid], red[tid + o]);
    __syncthreads();
  }
  const float M = red[0];
  __syncthreads();
  float sv = 0.0f;
  for (int w = tid; w < W; w += 256) sv = fmaf(sArr[w], expf(mArr[w] - M), sv);
  red[tid] = sv; __syncthreads();
  for (int o = 128; o > 0; o >>= 1) {
    if (tid < o) red[tid] += red[tid + o];
    __syncthreads();
  }
  if (tid == 0) { MS[0] = M; MS[1] = red[0]; }
}

// ---------------------------------------------------------------------------
// Kernel 2b: 32 blocks each reduce W/32 wave-partials (scaled by exp(m_w-M))
// ---------------------------------------------------------------------------
__global__ __launch_bounds__(256)
void uap_partial_kernel(const float* __restrict__ accA, const float* __restrict__ mArr,
                        const float* __restrict__ MS, float* __restrict__ accB, int W) {
  const int b   = blockIdx.x;
  const int per = W / 32;
  const int w0  = b * per;
  const float M = MS[0];
  const int tid = threadIdx.x;
  float s0 = 0.0f, s1 = 0.0f;
  for (int w = w0; w < w0 + per; ++w) {
    const float e = expf(mArr[w] - M);
    const float* row = accA + (size_t)w * DIM;
    s0 = fmaf(e, row[tid], s0);
    s1 = fmaf(e, row[tid + 256], s1);
  }
  accB[(size_t)b * DIM + tid]       = s0;
  accB[(size_t)b * DIM + tid + 256] = s1;
}

// ---------------------------------------------------------------------------
// Kernel 2c: final combine + normalize
// ---------------------------------------------------------------------------
__global__ __launch_bounds__(512)
void uap_final_kernel(const float* __restrict__ accB, const float* __restrict__ MS,
                      float* __restrict__ out) {
  const int d = threadIdx.x;
  float v = 0.0f;
#pragma unroll
  for (int b = 0; b < 32; ++b) v += accB[(size_t)b * DIM + d];
  out[d] = v / MS[1];
}

extern "C" void kernel_launch(void* const* d_in, const int* in_sizes, int n_in,
                              void* d_out, int out_size, void* d_ws, size_t ws_size,
                              hipStream_t stream) {
  const float* emb = (const float*)d_in[0];
  const float* q   = (const float*)d_in[1];
  float* out       = (float*)d_out;

  const int N      = in_sizes[0] / DIM;   // 200000
  const int nTiles = N / TILE;            // 12500

  // deterministic workspace sizing: W waves of partials
  int W = 4096;
  while (W > 256) {
    size_t need = sizeof(float) * ((size_t)W * DIM + 2 * (size_t)W + 2 + 32 * (size_t)DIM);
    if (need <= ws_size) break;
    W >>= 1;
  }

  float* accA = (float*)d_ws;
  float* mArr = accA + (size_t)W * DIM;
  float* sArr = mArr + W;
  float* MS   = sArr + W;
  float* accB = MS + 2;

  const int blocks = W / 8;  // 8 waves (256 threads) per block

  uap_stream_kernel<<<blocks, 256, 0, stream>>>(emb, q, accA, mArr, sArr, nTiles, W);
  uap_ms_kernel<<<1, 256, 0, stream>>>(mArr, sArr, MS, W);
  uap_partial_kernel<<<32, 256, 0, stream>>>(accA, mArr, MS, accB, W);
  uap_final_kernel<<<1, DIM, 0, stream>>>(accB, MS, out);
}